// VQLayer_9947144257863
// MI455X (gfx1250) — compile-verified
//
#include <hip/hip_runtime.h>

// MI455X / gfx1250, wave32. f32 WMMA path: V_WMMA_F32_16X16X4_F32.
typedef __attribute__((ext_vector_type(2))) float v2f;
typedef __attribute__((ext_vector_type(8))) float v8f;

#define B_ 32
#define F_ 256
#define T_ 4096
#define K_ 512
#define TM 32   // (b,t) rows per workgroup (two 16-row M tiles)

// ---------------------------------------------------------------------------
// Kernel 1: per-codebook-row squared norm + pack emb[K][F] into pair-major
// embP[f/2][k][2] so a WMMA B fragment is ONE global_load_b64 per lane.
// ---------------------------------------------------------------------------
__global__ __launch_bounds__(256) void vq_prep(const float* __restrict__ emb,
                                               float* __restrict__ csq,
                                               float* __restrict__ embP) {
  const int k = blockIdx.x;    // 0..511
  const int f = threadIdx.x;   // 0..255
  float v = emb[k * F_ + f];
  embP[((size_t)(f >> 1) * K_ + k) * 2 + (f & 1)] = v;
  __shared__ float red[256];
  red[f] = v * v;
  __syncthreads();
  #pragma unroll
  for (int s = 128; s > 0; s >>= 1) {
    if (f < s) red[f] += red[f + s];
    __syncthreads();
  }
  if (f == 0) csq[k] = red[0];
}

// ---------------------------------------------------------------------------
// Kernel 2: one block = 32 rows of (b,t) x all 512 codes.
// 8 waves; each wave: 64-wide K slice x 2 M tiles = 8 accumulators (64 VGPR).
// Inner f-step: 8 wmma per 4 global b64 (B) + 4 ds b32 (A).
// ---------------------------------------------------------------------------
__global__ __launch_bounds__(256) void vq_main(const float* __restrict__ x,
                                               const float* __restrict__ emb,
                                               const float* __restrict__ csq,
                                               const float* __restrict__ embP,
                                               float* __restrict__ quant,
                                               float* __restrict__ dist) {
  __shared__ float ldsX[2][F_ * 16 + 16];      // two 16-row halves, padded
  __shared__ float red[256];
  __shared__ float ldsXsq[TM];
  __shared__ unsigned long long ldsMin[TM];
  __shared__ int ldsIdx[TM];

  const int tid = threadIdx.x;
  const int r0  = blockIdx.x * TM;             // global row = b*T + t
  const int b   = r0 / T_;                     // T % TM == 0 -> same b
  const int t0  = r0 - b * T_;

  // Stage x tile [32 t][256 f] -> LDS [half][f][t&15]; accumulate squares.
  // Each thread keeps a fixed t = tid&31, f = tid>>5 + 8*j (coalesced 128B).
  const float* xb = x + (size_t)b * F_ * T_ + t0;
  {
    const int t = tid & 31;
    const int h = t >> 4;
    const int tl = t & 15;
    float psum = 0.f;
    #pragma unroll
    for (int j = 0; j < 32; ++j) {
      const int f = (tid >> 5) + j * 8;
      float v = xb[(size_t)f * T_ + t];
      ldsX[h][f * 16 + tl] = v;
      psum += v * v;
    }
    red[tid] = psum;
  }
  if (tid < TM) ldsMin[tid] = ~0ull;
  __syncthreads();
  if (tid < TM) {               // deterministic fixed-order combine
    float s = 0.f;
    #pragma unroll
    for (int sg = 0; sg < 8; ++sg) s += red[sg * 32 + tid];
    ldsXsq[tid] = s;
  }
  __syncthreads();

  const int lane  = tid & 31;
  const int wave  = tid >> 5;          // 0..7
  const int n     = lane & 15;         // N col within 16-tile / M row for A
  const int fsel  = (lane >> 4) << 1;  // lanes 0-15: K=f0,f0+1 ; 16-31: f0+2,f0+3
  const int kbase = wave * 64;
  const v2f* __restrict__ embPv = (const v2f*)embP;  // [F/2][K] of float2

  v8f acc[2][4] = {};

  for (int f0 = 0; f0 < F_; f0 += 4) {
    const int fr = f0 + fsel;
    v2f a0, a1;                               // A fragments for both M tiles
    a0.x = ldsX[0][fr * 16 + n];
    a0.y = ldsX[0][(fr + 1) * 16 + n];
    a1.x = ldsX[1][fr * 16 + n];
    a1.y = ldsX[1][(fr + 1) * 16 + n];
    const v2f* bp = embPv + (size_t)(fr >> 1) * K_ + kbase + n;
    #pragma unroll
    for (int kt = 0; kt < 4; ++kt) {
      v2f bf = bp[kt * 16];                   // one global_load_b64
      acc[0][kt] = __builtin_amdgcn_wmma_f32_16x16x4_f32(
          false, a0, false, bf, (short)0, acc[0][kt], false, false);
      acc[1][kt] = __builtin_amdgcn_wmma_f32_16x16x4_f32(
          false, a1, false, bf, (short)0, acc[1][kt], false, false);
    }
  }

  // Epilogue: dist = xsq - 2*cross + csq ; write dist ; argmin via ds_min_u64.
  const int mhi = (lane >> 4) << 3;    // C/D layout: VGPR j -> M=j / j+8
  #pragma unroll
  for (int mt = 0; mt < 2; ++mt) {
    #pragma unroll
    for (int kt = 0; kt < 4; ++kt) {
      const int k0 = kbase + kt * 16;
      const float cs = csq[k0 + n];
      #pragma unroll
      for (int j = 0; j < 8; ++j) {
        const int m = mt * 16 + j + mhi;      // row within 32-row strip
        float d = ldsXsq[m] - 2.0f * acc[mt][kt][j] + cs;
        dist[(size_t)(r0 + m) * K_ + (k0 + n)] = d;
        unsigned int u = __float_as_uint(d);
        u = (u & 0x80000000u) ? ~u : (u | 0x80000000u);
        unsigned long long key =
            ((unsigned long long)u << 32) | (unsigned int)(k0 + n);
        atomicMin(&ldsMin[m], key);           // lowest k wins ties
      }
    }
  }
  __syncthreads();
  if (tid < TM) ldsIdx[tid] = (int)(ldsMin[tid] & 0xFFFFFFFFull);
  __syncthreads();

  // Gather: quant[b, f, t0+t] = emb[idx[t], f]; 32 consecutive t -> 128B stores.
  {
    const int t   = tid & 31;
    const int fb  = tid >> 5;                 // 0..7
    const int idx = ldsIdx[t];
    const float* erow = emb + (size_t)idx * F_;
    float* qb = quant + (size_t)b * F_ * T_ + t0 + t;
    #pragma unroll
    for (int i = 0; i < 32; ++i) {
      const int f = fb + i * 8;
      qb[(size_t)f * T_] = erow[f];
    }
  }
}

extern "C" void kernel_launch(void* const* d_in, const int* in_sizes, int n_in,
                              void* d_out, int out_size, void* d_ws, size_t ws_size,
                              hipStream_t stream) {
  (void)in_sizes; (void)n_in; (void)out_size; (void)ws_size;
  const float* x   = (const float*)d_in[0];   // [B, F, T]
  const float* emb = (const float*)d_in[1];   // [K, F]
  float* quant = (float*)d_out;                               // [B, F, T]
  float* dist  = (float*)d_out + (size_t)B_ * F_ * T_;        // [B, T, K]
  float* csq   = (float*)d_ws;                                // 512 floats
  float* embP  = (float*)d_ws + K_;                           // 256*512 floats

  vq_prep<<<K_, 256, 0, stream>>>(emb, csq, embP);
  vq_main<<<(B_ * T_) / TM, 256, 0, stream>>>(x, emb, csq, embP, quant, dist);
}